// SpectralEmbedding_60670708023881
// MI455X (gfx1250) — compile-verified
//
#include <hip/hip_runtime.h>
#include <math.h>

typedef __attribute__((ext_vector_type(2))) float v2f;
typedef __attribute__((ext_vector_type(4))) float v4f;
typedef __attribute__((ext_vector_type(8))) float v8f;
typedef __attribute__((ext_vector_type(4))) int   v4i;

#define B_DIM   16
#define C_DIM   256
#define T_DIM   8192
#define SMALL_C 16
#define BINS    16
#define NFFT    30

// gfx1250 async global->LDS path (ASYNCcnt-tracked), with safe fallback
#if defined(__has_builtin)
#if __has_builtin(__builtin_amdgcn_global_load_async_to_lds_b128) && \
    __has_builtin(__builtin_amdgcn_s_wait_asynccnt)
#define USE_ASYNC_LDS 1
#endif
#endif
#ifndef USE_ASYNC_LDS
#define USE_ASYNC_LDS 0
#endif

#define AS1 __attribute__((address_space(1)))
#define AS3 __attribute__((address_space(3)))

static __device__ __forceinline__ v8f wmma_f32(v2f a, v2f b, v8f c) {
  // D = A(16x4) * B(4x16) + C, exact fp32 matrix op on CDNA5
  return __builtin_amdgcn_wmma_f32_16x16x4_f32(false, a, false, b, (short)0, c,
                                               false, false);
}

// ---------------------------------------------------------------------------
// Kernel 1: Gram_b = x_b * x_b^T   (256x256 per batch, K = T = 8192)
// grid (4,4,B), 128 threads (4 waves). WG computes one 64x64 tile.
// Wave w computes rows [i0+16w, +16) x cols [j0, j0+64) = 1x4 WMMA tiles.
// LDS is double-buffered; chunk i+1 streams in (async if available) while
// chunk i feeds the WMMA pipe.
// ---------------------------------------------------------------------------
#define TK   64
#define LDSS 68   // 64 + 4 pad: conflict-free 8B fragment reads

__global__ __launch_bounds__(128) void gram_kernel(const float* __restrict__ x,
                                                   float* __restrict__ G) {
  __shared__ __align__(16) float As[2][64 * LDSS];
  __shared__ __align__(16) float Bs[2][64 * LDSS];

  const int b    = blockIdx.z;
  const int i0   = blockIdx.y * 64;
  const int j0   = blockIdx.x * 64;
  const int tid  = threadIdx.x;
  const int w    = tid >> 5;        // wave 0..3
  const int lane = tid & 31;
  const int m    = lane & 15;       // row within 16 (A: M, B: N)
  const int kh   = lane >> 4;       // K half-select (0 -> K0,1 ; 1 -> K2,3)

  const int c4 = (tid & 15) * 4;    // column (t) within chunk, float4 granule
  const int rb = tid >> 4;          // 0..7 row base for staging

  // issue one 64x64 f32 chunk (A rows i0.., B rows j0..) into buffer `buf`
  auto issue = [&](int chunk, int buf) {
    const int t0 = chunk * TK;
#pragma unroll
    for (int rr = 0; rr < 8; ++rr) {
      const int r = rb + rr * 8;
      const float* ga = x + ((size_t)(b * C_DIM + i0 + r)) * T_DIM + t0 + c4;
      const float* gb = x + ((size_t)(b * C_DIM + j0 + r)) * T_DIM + t0 + c4;
#if USE_ASYNC_LDS
      __builtin_amdgcn_global_load_async_to_lds_b128(
          (AS1 v4i*)ga, (AS3 v4i*)(&As[buf][r * LDSS + c4]), 0, 0);
      __builtin_amdgcn_global_load_async_to_lds_b128(
          (AS1 v4i*)gb, (AS3 v4i*)(&Bs[buf][r * LDSS + c4]), 0, 0);
#else
      *(v4f*)(&As[buf][r * LDSS + c4]) = *(const v4f*)ga;
      *(v4f*)(&Bs[buf][r * LDSS + c4]) = *(const v4f*)gb;
#endif
    }
  };

  v8f acc[4] = {};

  const int NCH = T_DIM / TK;  // 128 chunks
  issue(0, 0);
  for (int i = 0; i < NCH; ++i) {
    const int cur = i & 1;
    if (i + 1 < NCH) {
      issue(i + 1, cur ^ 1);           // prefetch next chunk (16 xfers/thread)
#if USE_ASYNC_LDS
      __builtin_amdgcn_s_wait_asynccnt(16);  // chunk i's 16 xfers retired
#endif
    } else {
#if USE_ASYNC_LDS
      __builtin_amdgcn_s_wait_asynccnt(0);
#endif
    }
    __syncthreads();                    // chunk i visible to all waves

    const float* Ac = &As[cur][0];
    const float* Bc = &Bs[cur][0];
#pragma unroll
    for (int kk = 0; kk < TK / 4; ++kk) {
      const int ko = kk * 4 + kh * 2;
      v2f a = *(const v2f*)(&Ac[(w * 16 + m) * LDSS + ko]);
#pragma unroll
      for (int tj = 0; tj < 4; ++tj) {
        v2f bb = *(const v2f*)(&Bc[(tj * 16 + m) * LDSS + ko]);
        acc[tj] = wmma_f32(a, bb, acc[tj]);
      }
    }
    __syncthreads();                    // done reading buf before overwrite
  }

  // C/D layout: VGPR r, lanes 0-15 -> M=r, lanes 16-31 -> M=r+8; N = lane%16
#pragma unroll
  for (int tj = 0; tj < 4; ++tj) {
#pragma unroll
    for (int r = 0; r < 8; ++r) {
      const int row = i0 + w * 16 + r + 8 * kh;
      const int col = j0 + tj * 16 + m;
      G[((size_t)b * C_DIM + row) * C_DIM + col] = acc[tj][r];
    }
  }
}

// ---------------------------------------------------------------------------
// Kernel 2: S_b = softmax(Wk * G_b * Wq^T / 16) ;  M_b = S_b * Wv
// grid (B), 256 threads. Tiny (100 MFLOP total) -> VALU.
// ---------------------------------------------------------------------------
__global__ __launch_bounds__(256) void attn_kernel(const float* __restrict__ G,
                                                   const float* __restrict__ Wk,
                                                   const float* __restrict__ Wq,
                                                   const float* __restrict__ Wv,
                                                   float* __restrict__ Mm) {
  __shared__ float Yl[SMALL_C * C_DIM];
  __shared__ float Sl[SMALL_C * C_DIM];
  const int b = blockIdx.x;
  const int t = threadIdx.x;  // 0..255

  // Y[i][t] = sum_C Wk[i][C] * G[b][C][t]
  float y[SMALL_C];
#pragma unroll
  for (int i = 0; i < SMALL_C; ++i) y[i] = 0.f;
  for (int Cc = 0; Cc < C_DIM; ++Cc) {
    const float g = G[((size_t)b * C_DIM + Cc) * C_DIM + t];
#pragma unroll
    for (int i = 0; i < SMALL_C; ++i) y[i] = fmaf(Wk[i * C_DIM + Cc], g, y[i]);
  }
#pragma unroll
  for (int i = 0; i < SMALL_C; ++i) Yl[i * C_DIM + t] = y[i];
  __syncthreads();

  // S[i][j=t] = (1/16) * sum_C' Y[i][C'] * Wq[j][C']
  float s[SMALL_C];
#pragma unroll
  for (int i = 0; i < SMALL_C; ++i) s[i] = 0.f;
  for (int Cc = 0; Cc < C_DIM; ++Cc) {
    const float wq = Wq[(size_t)t * C_DIM + Cc];
#pragma unroll
    for (int i = 0; i < SMALL_C; ++i) s[i] = fmaf(Yl[i * C_DIM + Cc], wq, s[i]);
  }
#pragma unroll
  for (int i = 0; i < SMALL_C; ++i) Sl[i * C_DIM + t] = s[i] * (1.0f / 16.0f);
  __syncthreads();

  // softmax over j for each of the 16 rows (threads 0..15)
  if (t < SMALL_C) {
    float mx = -3.402823466e38f;
    for (int j = 0; j < C_DIM; ++j) mx = fmaxf(mx, Sl[t * C_DIM + j]);
    float sum = 0.f;
    for (int j = 0; j < C_DIM; ++j) {
      const float e = __expf(Sl[t * C_DIM + j] - mx);
      Sl[t * C_DIM + j] = e;
      sum += e;
    }
    const float inv = 1.0f / sum;
    for (int j = 0; j < C_DIM; ++j) Sl[t * C_DIM + j] *= inv;
  }
  __syncthreads();

  // M[i][t] = sum_j S[i][j] * Wv[j][t]
  float mm[SMALL_C];
#pragma unroll
  for (int i = 0; i < SMALL_C; ++i) mm[i] = 0.f;
  for (int j = 0; j < C_DIM; ++j) {
    const float wv = Wv[(size_t)j * C_DIM + t];
#pragma unroll
    for (int i = 0; i < SMALL_C; ++i) mm[i] = fmaf(Sl[i * C_DIM + j], wv, mm[i]);
  }
#pragma unroll
  for (int i = 0; i < SMALL_C; ++i)
    Mm[((size_t)b * SMALL_C + i) * C_DIM + t] = mm[i];
}

// ---------------------------------------------------------------------------
// Kernel 3: out_b[16,T] = M_b[16,256] * x_b[256,T]   (WMMA f32, K=256)
// grid (T/64, B), 128 threads (4 waves), one 16x16 t-tile per wave.
// ---------------------------------------------------------------------------
__global__ __launch_bounds__(128) void outproj_kernel(const float* __restrict__ x,
                                                      const float* __restrict__ Mm,
                                                      float* __restrict__ sig) {
  const int b    = blockIdx.y;
  const int tid  = threadIdx.x;
  const int w    = tid >> 5;
  const int lane = tid & 31;
  const int m    = lane & 15;
  const int kh   = lane >> 4;
  const int t    = blockIdx.x * 64 + w * 16 + m;  // N index

  v8f acc = {};
  const float* Mb = Mm + (size_t)b * SMALL_C * C_DIM;
#pragma unroll 4
  for (int k0 = 0; k0 < C_DIM; k0 += 4) {
    const int kr = k0 + kh * 2;
    v2f a = *(const v2f*)(Mb + m * C_DIM + kr);  // A: M row m, K kr..kr+1
    v2f bb;                                      // B: K rows of x at column t
    bb.x = x[((size_t)(b * C_DIM + kr)) * T_DIM + t];
    bb.y = x[((size_t)(b * C_DIM + kr + 1)) * T_DIM + t];
    acc = wmma_f32(a, bb, acc);
  }
#pragma unroll
  for (int r = 0; r < 8; ++r)
    sig[((size_t)(b * SMALL_C) + r + 8 * kh) * T_DIM + t] = acc[r];
}

// ---------------------------------------------------------------------------
// Kernel 4: power spectrogram (n_fft=30, hop=1, hann, center/reflect, /sum(w^2))
// grid (T/256, B*16), 256 threads; signal slice + DFT tables in LDS.
// Output layout: out[(nsig*BINS + k)*T + t]  (== reshape(B, 256, T))
// ---------------------------------------------------------------------------
__global__ __launch_bounds__(256) void spect_kernel(const float* __restrict__ sig,
                                                    float* __restrict__ out) {
  __shared__ float sh_sig[256 + NFFT];       // 285 used
  __shared__ float ct[BINS * NFFT];
  __shared__ float st[BINS * NFFT];

  const int nsig = blockIdx.y;
  const int t0   = blockIdx.x * 256;
  const int tid  = threadIdx.x;
  const float TWO_PI = 6.283185307179586f;

  // window * DFT tables
  for (int idx = tid; idx < BINS * NFFT; idx += 256) {
    const int k = idx / NFFT, n = idx - k * NFFT;
    const float wdw = 0.5f - 0.5f * cosf(TWO_PI * (float)n / (float)NFFT);
    const float ang = TWO_PI * (float)(k * n) / (float)NFFT;
    ct[idx] = cosf(ang) * wdw;
    st[idx] = sinf(ang) * wdw;
  }
  // signal slice with reflect padding (pad = 15)
  for (int idx = tid; idx < 256 + NFFT - 1; idx += 256) {
    int gi = t0 - (NFFT / 2) + idx;
    if (gi < 0) gi = -gi;
    if (gi >= T_DIM) gi = 2 * T_DIM - 2 - gi;
    sh_sig[idx] = sig[(size_t)nsig * T_DIM + gi];
  }
  // normalization 1/sum(w^2)
  float wsum = 0.f;
  for (int n = 0; n < NFFT; ++n) {
    const float wdw = 0.5f - 0.5f * cosf(TWO_PI * (float)n / (float)NFFT);
    wsum = fmaf(wdw, wdw, wsum);
  }
  const float norm = 1.0f / wsum;
  __syncthreads();

  float re[BINS], im[BINS];
#pragma unroll
  for (int k = 0; k < BINS; ++k) { re[k] = 0.f; im[k] = 0.f; }
  for (int n = 0; n < NFFT; ++n) {
    const float s = sh_sig[tid + n];
#pragma unroll
    for (int k = 0; k < BINS; ++k) {
      re[k] = fmaf(s, ct[k * NFFT + n], re[k]);  // LDS broadcast reads
      im[k] = fmaf(s, st[k * NFFT + n], im[k]);
    }
  }
  const int t = t0 + tid;
#pragma unroll
  for (int k = 0; k < BINS; ++k)
    out[((size_t)nsig * BINS + k) * T_DIM + t] =
        (re[k] * re[k] + im[k] * im[k]) * norm;
}

// ---------------------------------------------------------------------------
extern "C" void kernel_launch(void* const* d_in, const int* in_sizes, int n_in,
                              void* d_out, int out_size, void* d_ws, size_t ws_size,
                              hipStream_t stream) {
  const float* x  = (const float*)d_in[0];
  const float* Wk = (const float*)d_in[1];
  const float* Wq = (const float*)d_in[2];
  const float* Wv = (const float*)d_in[3];
  float* out = (float*)d_out;

  // workspace: Gram [B,256,256] | M [B,16,256] | out-signal [B*16, T]  (~12.25 MB)
  float* G   = (float*)d_ws;
  float* Mm  = G + (size_t)B_DIM * C_DIM * C_DIM;
  float* sg  = Mm + (size_t)B_DIM * SMALL_C * C_DIM;

  gram_kernel<<<dim3(4, 4, B_DIM), 128, 0, stream>>>(x, G);
  attn_kernel<<<dim3(B_DIM), 256, 0, stream>>>(G, Wk, Wq, Wv, Mm);
  outproj_kernel<<<dim3(T_DIM / 64, B_DIM), 128, 0, stream>>>(x, Mm, sg);
  spect_kernel<<<dim3(T_DIM / 256, B_DIM * SMALL_C), 256, 0, stream>>>(sg, out);
}